// FinePreprocess_61426622267544
// MI455X (gfx1250) — compile-verified
//
#include <hip/hip_runtime.h>
#include <hip/hip_bf16.h>

typedef __attribute__((ext_vector_type(2))) float v2f;
typedef __attribute__((ext_vector_type(8))) float v8f;
typedef int v4i __attribute__((vector_size(16)));  // matches builtin param type

#define N_MATCH 16384
#define D_F 64
#define D_C 256
#define H_F 240
#define W_F 320
#define L_C 4800
#define WW 25

// ---------------------------------------------------------------------------
// Async global->LDS copy helpers (gfx1250 ASYNCcnt path) with sync fallback.
// ---------------------------------------------------------------------------
#if defined(__gfx1250__) && __has_builtin(__builtin_amdgcn_global_load_async_to_lds_b32)
#define HAS_ASYNC_LDS 1
#else
#define HAS_ASYNC_LDS 0
#endif

typedef __attribute__((address_space(1))) int* as1_i32p;
typedef __attribute__((address_space(3))) int* as3_i32p;
typedef __attribute__((address_space(1))) v4i* as1_v4ip;
typedef __attribute__((address_space(3))) v4i* as3_v4ip;

__device__ __forceinline__ void copy_b32_to_lds(const float* __restrict__ gsrc,
                                                float* lds_dst) {
#if HAS_ASYNC_LDS
  __builtin_amdgcn_global_load_async_to_lds_b32((as1_i32p)gsrc, (as3_i32p)lds_dst,
                                                0, 0);
#else
  *lds_dst = *gsrc;
#endif
}

__device__ __forceinline__ void copy_b128_to_lds(const float* __restrict__ gsrc,
                                                 float* lds_dst) {
#if HAS_ASYNC_LDS && __has_builtin(__builtin_amdgcn_global_load_async_to_lds_b128)
  __builtin_amdgcn_global_load_async_to_lds_b128((as1_v4ip)gsrc, (as3_v4ip)lds_dst,
                                                 0, 0);
#else
  *(float4*)lds_dst = *(const float4*)gsrc;
#endif
}

__device__ __forceinline__ void async_copy_wait() {
#if HAS_ASYNC_LDS
#if __has_builtin(__builtin_amdgcn_s_wait_asynccnt)
  __builtin_amdgcn_s_wait_asynccnt(0);
#else
  asm volatile("s_wait_asynccnt 0x0" ::: "memory");
#endif
#endif
}

// ---------------------------------------------------------------------------
// Kernel A: Wfused = Wd @ Wm[64:128], bfused = bd @ Wm[64:128] + bm
// Tiny (2 MFLOP) — plain VALU.
// ---------------------------------------------------------------------------
__global__ __launch_bounds__(256) void fuse_weights(
    const float* __restrict__ Wd, const float* __restrict__ bd,
    const float* __restrict__ Wm, const float* __restrict__ bm,
    float* __restrict__ Wfused, float* __restrict__ bfused) {
  int idx = blockIdx.x * blockDim.x + threadIdx.x;  // 0..16383
  int i = idx >> 6;                                 // 0..255 (row of Wd)
  int j = idx & 63;                                 // 0..63  (col)
  float acc = 0.f;
#pragma unroll 8
  for (int k = 0; k < 64; ++k)
    acc += Wd[i * 64 + k] * Wm[(64 + k) * 64 + j];
  Wfused[idx] = acc;
  if (idx < 64) {
    float b = bm[idx];
    for (int k = 0; k < 64; ++k)
      b += bd[k] * Wm[(64 + k) * 64 + idx];
    bfused[idx] = b;
  }
}

// ---------------------------------------------------------------------------
// Kernel B: ccontrib[2N][64] = gather(c rows) @ Wfused + bfused
// M-tile = 16 gathered rows per block, 4 waves cover 4 N-tiles of 16 cols.
// K = 256 -> 64 V_WMMA_F32_16X16X4_F32 per tile. Rows staged via async LDS.
// ---------------------------------------------------------------------------
__global__ __launch_bounds__(128) void coarse_contrib(
    const float* __restrict__ feat_c0, const float* __restrict__ feat_c1,
    const int* __restrict__ b_ids, const int* __restrict__ ci_ids,
    const float* __restrict__ Wfused, const float* __restrict__ bfused,
    float* __restrict__ ccontrib) {
  __shared__ float A[16 * 260];  // padded stride 260 -> bank-conflict-free
  const int tid = threadIdx.x;
  const int rowbase = blockIdx.x * 16;

  // Gather 16 coarse rows (256 contiguous floats each) into LDS, 16B chunks.
  {
    int r = tid >> 3;  // row 0..15
    int q = tid & 7;   // 8 threads per row
    int gr = rowbase + r;
    int n = (gr < N_MATCH) ? gr : gr - N_MATCH;
    const float* src = (gr < N_MATCH) ? feat_c0 : feat_c1;
    const size_t off = ((size_t)b_ids[n] * L_C + (size_t)ci_ids[n]) * D_C;
    const float* srow = src + off;
#pragma unroll
    for (int j = 0; j < 8; ++j) {
      int e = 4 * (q + j * 8);  // 16B-aligned in both global and padded LDS
      copy_b128_to_lds(srow + e, &A[r * 260 + e]);
    }
  }
  async_copy_wait();
  __syncthreads();

  const int lane = tid & 31;
  const int wv = tid >> 5;                // N-tile id 0..3
  const int col = wv * 16 + (lane & 15);  // output column
  const int koff = (lane >> 4) * 2;       // WMMA f32 A/B lane->K mapping
  const int m = lane & 15;

  v8f acc = {};
#pragma unroll 8
  for (int ks = 0; ks < 64; ++ks) {
    int k = ks * 4 + koff;
    v2f a, b;
    a.x = A[m * 260 + k];
    a.y = A[m * 260 + k + 1];
    b.x = Wfused[(k) * 64 + col];  // coalesced, L2-hot
    b.y = Wfused[(k + 1) * 64 + col];
    acc = __builtin_amdgcn_wmma_f32_16x16x4_f32(false, a, false, b,
                                                (short)0, acc, false, false);
  }
  const float bias = bfused[col];
#pragma unroll
  for (int i = 0; i < 8; ++i) {
    int row = rowbase + ((lane < 16) ? i : i + 8);
    ccontrib[(size_t)row * 64 + col] = acc[i] + bias;
  }
}

// ---------------------------------------------------------------------------
// Kernel C: per (source s, match n): out = f_win(25x64) @ Wm_top(64x64) + cc
// M padded 25->32 (two 16-row WMMA tiles), 4 waves = 4 N-tiles of 16 cols.
// Window gathered via async-to-LDS; B fragments preloaded in registers.
// ---------------------------------------------------------------------------
__global__ __launch_bounds__(128) void fine_merge(
    const float* __restrict__ feat_f0, const float* __restrict__ feat_f1,
    const int* __restrict__ b_ids, const int* __restrict__ cy,
    const int* __restrict__ cx, const float* __restrict__ Wm,
    const float* __restrict__ ccontrib, float* __restrict__ out) {
  __shared__ float A[32 * 68];  // padded stride 68 -> bank-conflict-free
  const int n = blockIdx.x;
  const int s = blockIdx.y;
  const int tid = threadIdx.x;
  const float* feat = s ? feat_f1 : feat_f0;
  const int b = b_ids[n];
  const int y0 = cy[n] - 2;
  const int x0 = cx[n] - 2;

  // Zero the pad rows (window pixels 25..31).
  for (int i = tid; i < 7 * 64; i += 128) {
    int w = 25 + (i >> 6);
    A[w * 68 + (i & 63)] = 0.f;
  }
  // Gather window: 64 channels x 5 rows = 320 segments of 5 contiguous floats.
  for (int seg = tid; seg < 320; seg += 128) {
    int k = seg / 5;
    int dy = seg - k * 5;
    const float* src = feat + (((size_t)b * D_F + k) * H_F + (y0 + dy)) * W_F + x0;
#pragma unroll
    for (int dx = 0; dx < 5; ++dx)
      copy_b32_to_lds(src + dx, &A[(dy * 5 + dx) * 68 + k]);
  }
  async_copy_wait();
  __syncthreads();

  const int lane = tid & 31;
  const int wv = tid >> 5;
  const int col = wv * 16 + (lane & 15);
  const int koff = (lane >> 4) * 2;

  // Preload all B fragments of Wm_top (rows 0..63) for this wave's N-tile.
  v2f bfrag[16];
#pragma unroll
  for (int ks = 0; ks < 16; ++ks) {
    int k = ks * 4 + koff;
    bfrag[ks].x = Wm[(k) * 64 + col];
    bfrag[ks].y = Wm[(k + 1) * 64 + col];
  }
  const float cc = ccontrib[((size_t)s * N_MATCH + n) * 64 + col];
  const size_t obase = ((size_t)s * N_MATCH + n) * WW * 64;

#pragma unroll
  for (int mt = 0; mt < 2; ++mt) {
    const int m = mt * 16 + (lane & 15);
    v8f acc = {};
#pragma unroll
    for (int ks = 0; ks < 16; ++ks) {
      int k = ks * 4 + koff;
      v2f a;
      a.x = A[m * 68 + k];
      a.y = A[m * 68 + k + 1];
      acc = __builtin_amdgcn_wmma_f32_16x16x4_f32(false, a, false, bfrag[ks],
                                                  (short)0, acc, false, false);
    }
#pragma unroll
    for (int i = 0; i < 8; ++i) {
      int w = mt * 16 + ((lane < 16) ? i : i + 8);
      if (w < WW)
        out[obase + (size_t)w * 64 + col] = acc[i] + cc;
    }
  }
}

// ---------------------------------------------------------------------------
extern "C" void kernel_launch(void* const* d_in, const int* in_sizes, int n_in,
                              void* d_out, int out_size, void* d_ws, size_t ws_size,
                              hipStream_t stream) {
  const float* feat_f0 = (const float*)d_in[0];
  const float* feat_f1 = (const float*)d_in[1];
  const float* feat_c0 = (const float*)d_in[2];
  const float* feat_c1 = (const float*)d_in[3];
  const float* Wd = (const float*)d_in[4];
  const float* bd = (const float*)d_in[5];
  const float* Wm = (const float*)d_in[6];
  const float* bm = (const float*)d_in[7];
  const int* b_ids = (const int*)d_in[8];
  const int* cy = (const int*)d_in[9];
  const int* cx = (const int*)d_in[10];
  const int* ci = (const int*)d_in[11];
  float* out = (float*)d_out;

  float* Wfused = (float*)d_ws;       // 256*64 floats
  float* bfused = Wfused + 256 * 64;  // 64 floats
  float* ccontrib = bfused + 64;      // 32768*64 floats (~8 MB)

  fuse_weights<<<64, 256, 0, stream>>>(Wd, bd, Wm, bm, Wfused, bfused);
  coarse_contrib<<<(2 * N_MATCH) / 16, 128, 0, stream>>>(
      feat_c0, feat_c1, b_ids, ci, Wfused, bfused, ccontrib);
  fine_merge<<<dim3(N_MATCH, 2), 128, 0, stream>>>(
      feat_f0, feat_f1, b_ids, cy, cx, Wm, ccontrib, out);
}